// SimplePointRepulsionLoss_1382979470111
// MI455X (gfx1250) — compile-verified
//
#include <hip/hip_runtime.h>
#include <hip/hip_bf16.h>
#include <math.h>

typedef float v2f __attribute__((ext_vector_type(2)));
typedef float v8f __attribute__((ext_vector_type(8)));

#define BATCH   4
#define NPTS    8192
#define NN      16
#define KSLOTS  (NN + 1)     // keep self, drop it at the end
#define ROWS    256          // query rows per workgroup
#define CTILE   32           // candidate columns per iteration
#define THREADS 256          // 8 waves (wave32)

__launch_bounds__(THREADS)
__global__ void knn_repulsion_kernel(const float* __restrict__ points,
                                     float* __restrict__ partial) {
  const int b       = blockIdx.y;
  const int rowbase = blockIdx.x * ROWS;
  const int t       = threadIdx.x;
  const int lane    = t & 31;          // wave32
  const int wave    = t >> 5;
  const int hi      = (lane >> 4);     // 0: lanes 0-15, 1: lanes 16-31
  const int l16     = lane & 15;

  __shared__ float4 cpts[CTILE];            // candidate x,y,z,sq
  __shared__ float  dtile[ROWS * CTILE];    // 32 KB distance tile
  __shared__ float  red[THREADS];

  const float* pb = points + (size_t)b * NPTS * 3;

  // ---- Per-wave A operands: 2 row-tiles of 16 query rows each.
  // A row n = (-2x, -2y, -2z, sq_n).  16x4 f32 A layout:
  //   lanes 0-15 : VGPR0=K0, VGPR1=K1 (M=lane)
  //   lanes 16-31: VGPR0=K2, VGPR1=K3 (M=lane-16)
  v2f amat[2];
  #pragma unroll
  for (int rt = 0; rt < 2; ++rt) {
    int r = rowbase + (wave * 2 + rt) * 16 + l16;
    float px = pb[r * 3 + 0];
    float py = pb[r * 3 + 1];
    float pz = pb[r * 3 + 2];
    float sq = px * px + py * py + pz * pz;
    amat[rt].x = hi ? (-2.0f * pz) : (-2.0f * px);
    amat[rt].y = hi ? sq           : (-2.0f * py);
  }

  // Sorted ascending top-17 (register resident, all indexing unrolled).
  float s[KSLOTS];
  #pragma unroll
  for (int i = 0; i < KSLOTS; ++i) s[i] = 3.0e38f;

  for (int cbase = 0; cbase < NPTS; cbase += CTILE) {
    __syncthreads();                    // previous iteration's readers done
    if (t < CTILE) {
      int c = cbase + t;
      float x = pb[c * 3 + 0];
      float y = pb[c * 3 + 1];
      float z = pb[c * 3 + 2];
      cpts[t] = make_float4(x, y, z, x * x + y * y + z * z);
    }
    if (t == 0 && cbase + CTILE < NPTS) {
      __builtin_prefetch(pb + (size_t)(cbase + CTILE) * 3, 0, 1);
    }
    __syncthreads();

    // ---- WMMA phase: wave computes 2 row-tiles x 2 col-tiles of d^2 ----
    #pragma unroll
    for (int rt = 0; rt < 2; ++rt) {
      const int rlocal = (wave * 2 + rt) * 16;
      #pragma unroll
      for (int ct = 0; ct < CTILE / 16; ++ct) {
        float4 c4 = cpts[ct * 16 + l16];
        // B col m = (x, y, z, 1).  4x16 B layout (rows striped over halves):
        //   VGPR0: lanes0-15 = K0 row, lanes16-31 = K1 row
        //   VGPR1: lanes0-15 = K2 row, lanes16-31 = K3 row
        v2f bmat;
        bmat.x = hi ? c4.y : c4.x;
        bmat.y = hi ? 1.0f : c4.z;
        v8f cacc;
        #pragma unroll
        for (int j = 0; j < 8; ++j) cacc[j] = c4.w;   // + sq_m per column
        // D = A*B + C  =  sq_n + sq_m - 2*<p_n,p_m>  (exact d^2 in one op)
        v8f dres = __builtin_amdgcn_wmma_f32_16x16x4_f32(
            false, amat[rt], false, bmat, (short)0, cacc, false, false);
        #pragma unroll
        for (int j = 0; j < 8; ++j) {
          int row = rlocal + j + hi * 8;              // C/D layout: M=j(+8)
          dtile[row * CTILE + ct * 16 + l16] = dres[j];
        }
      }
    }
    __syncthreads();

    // ---- Selection: thread t owns query row t ----
    // Stream the row as float4 (ds_load_b128) and pre-screen each group of
    // 4 candidates with one compare; fall into the guarded bubble only when
    // some element beats the current 17th-smallest (rare: ~k*ln(N/k)/N).
    const float4* drow4 = (const float4*)&dtile[t * CTILE];
    #pragma unroll
    for (int g = 0; g < CTILE / 4; ++g) {
      float4 q = drow4[g];
      float m01 = fminf(q.x, q.y);
      float m23 = fminf(q.z, q.w);
      if (fminf(m01, m23) < s[KSLOTS - 1]) {
        #pragma unroll
        for (int e = 0; e < 4; ++e) {
          float d2 = (e == 0) ? q.x : (e == 1) ? q.y : (e == 2) ? q.z : q.w;
          if (d2 < s[KSLOTS - 1]) {
            s[KSLOTS - 1] = d2;
            #pragma unroll
            for (int i = KSLOTS - 1; i > 0; --i) {   // single bubble pass
              float a  = s[i - 1], bb = s[i];
              s[i - 1] = fminf(a, bb);
              s[i]     = fmaxf(a, bb);
            }
          }
        }
      }
    }
  }

  // s[0] ~ self distance (0); neighbors are s[1..16]
  float acc = 0.0f;
  #pragma unroll
  for (int i = 1; i < KSLOTS; ++i) {
    float d2 = fmaxf(s[i], 0.0f);
    if (d2 < 0.25f) acc += __frsqrt_rn(d2 + 1.0e-4f);
  }

  // Deterministic per-workgroup reduction -> partial sums in workspace
  red[t] = acc;
  __syncthreads();
  #pragma unroll
  for (int off = THREADS / 2; off > 0; off >>= 1) {
    if (t < off) red[t] += red[t + off];
    __syncthreads();
  }
  if (t == 0) partial[blockIdx.y * gridDim.x + blockIdx.x] = red[0];
}

__global__ void finalize_kernel(const float* __restrict__ partial,
                                float* __restrict__ out, int nPartial) {
  __shared__ float red[128];
  int t = threadIdx.x;
  red[t] = (t < nPartial) ? partial[t] : 0.0f;
  __syncthreads();
  #pragma unroll
  for (int off = 64; off > 0; off >>= 1) {
    if (t < off) red[t] += red[t + off];
    __syncthreads();
  }
  if (t == 0) out[0] = red[0] * (1.0f / ((float)BATCH * NPTS * NN));
}

extern "C" void kernel_launch(void* const* d_in, const int* in_sizes, int n_in,
                              void* d_out, int out_size, void* d_ws, size_t ws_size,
                              hipStream_t stream) {
  const float* points = (const float*)d_in[0];
  float* partial = (float*)d_ws;                 // 128 floats of scratch
  dim3 grid(NPTS / ROWS, BATCH);                 // 32 x 4 = 128 workgroups
  knn_repulsion_kernel<<<grid, THREADS, 0, stream>>>(points, partial);
  finalize_kernel<<<1, 128, 0, stream>>>(partial, (float*)d_out,
                                         (NPTS / ROWS) * BATCH);
}